// GraphVAELoss_25116968747265
// MI455X (gfx1250) — compile-verified
//
#include <hip/hip_runtime.h>

#define B_   32
#define N_   1024
#define ZI_  64
#define ZA_  128
#define SPW  0.7f
#define EPSV 1e-10f

typedef float v2f __attribute__((ext_vector_type(2)));
typedef float v4f __attribute__((ext_vector_type(4)));
typedef float v8f __attribute__((ext_vector_type(8)));

// ---------------- workspace init: ws[0]=adj sum, ws[1]=KL_A sum, ws[2]=KL_i sum
__global__ void init_ws_kernel(float* __restrict__ ws) {
    if (threadIdx.x < 3) ws[threadIdx.x] = 0.0f;
}

// ---------------- block reduction helper (wave32)
__device__ __forceinline__ float blockReduceSum(float v) {
    __shared__ float sm[32];
    const int lane = threadIdx.x & 31;
    const int w    = threadIdx.x >> 5;
    #pragma unroll
    for (int off = 16; off > 0; off >>= 1) v += __shfl_down(v, off, 32);
    if (lane == 0) sm[w] = v;
    __syncthreads();
    const int nw = blockDim.x >> 5;
    v = (threadIdx.x < nw) ? sm[threadIdx.x] : 0.0f;
    if (w == 0) {
        #pragma unroll
        for (int off = 16; off > 0; off >>= 1) v += __shfl_down(v, off, 32);
    }
    return v; // valid on thread 0
}

// ---------------- 1) adjacency reconstruction log-likelihood (streaming, 268 MB)
// Single-use stream > L2 capacity -> non-temporal loads (TH_NT) to avoid
// evicting reusable lines.
__global__ __launch_bounds__(256)
void adj_recon_kernel(const float* __restrict__ A_hat,
                      const float* __restrict__ A_true,
                      float* __restrict__ ws) {
    const long long total4 = (long long)B_ * N_ * N_ / 4;
    float acc = 0.0f;
    for (long long i = blockIdx.x * (long long)blockDim.x + threadIdx.x;
         i < total4; i += (long long)gridDim.x * blockDim.x) {
        const v4f a = __builtin_nontemporal_load((const v4f*)A_hat + i);
        const v4f t = __builtin_nontemporal_load((const v4f*)A_true + i);
        #pragma unroll
        for (int j = 0; j < 4; ++j) {
            float x = a[j];
            x = (x == 0.0f) ? EPSV : ((x == 1.0f) ? 1.0f - EPSV : x);
            // A_true is exactly 0.0 or 1.0:
            //   t==1: ll = log(A)          (no sparsity weight)
            //   t==0: ll = 0.7 * log(1-A)
            const bool one = (t[j] != 0.0f);
            const float arg = one ? x : (1.0f - x);
            const float wgt = one ? 1.0f : SPW;
            acc += wgt * __logf(arg);                  // v_log_f32
        }
    }
    const float s = blockReduceSum(acc);
    if (threadIdx.x == 0) atomicAdd(ws + 0, s);
}

// ---------------- 2) KL_A vs standard normal (tiny)
__global__ __launch_bounds__(128)
void kl_a_kernel(const float* __restrict__ mu_A,
                 const float* __restrict__ logvar_A,
                 float* __restrict__ ws) {
    const int b = blockIdx.x;
    const float m  = mu_A[b * ZA_ + threadIdx.x];
    const float lv = logvar_A[b * ZA_ + threadIdx.x];
    float s0 = lv;          // sum logvar
    float s1 = __expf(lv);  // trace term
    float s2 = m * m;       // inner product term (lv2 = 0)
    __shared__ float sm[3][4];
    const int lane = threadIdx.x & 31;
    const int w    = threadIdx.x >> 5;
    #pragma unroll
    for (int off = 16; off > 0; off >>= 1) {
        s0 += __shfl_down(s0, off, 32);
        s1 += __shfl_down(s1, off, 32);
        s2 += __shfl_down(s2, off, 32);
    }
    if (lane == 0) { sm[0][w] = s0; sm[1][w] = s1; sm[2][w] = s2; }
    __syncthreads();
    if (threadIdx.x == 0) {
        float t0 = 0.f, t1 = 0.f, t2 = 0.f;
        #pragma unroll
        for (int i = 0; i < 4; ++i) { t0 += sm[0][i]; t1 += sm[1][i]; t2 += sm[2][i]; }
        const float kl = 0.5f * __expf(t0) - (float)ZA_ + t1 + t2;
        atomicAdd(ws + 1, kl);
    }
}

// ---------------- 3) KL_i: per-node WMMA GEMM (64x128 @ 128x32) + KL epilogue
// One wave per node n. Grid is exactly 1024 waves -> EXEC all-ones at WMMA.
// B-matrix rows are single-use per wave -> non-temporal; z_a (16 KB) is
// reused by every wave -> regular temporal so it stays cache-resident.
__global__ __launch_bounds__(256)
void kl_i_kernel(const float* __restrict__ mu_i,
                 const float* __restrict__ logvar_i,
                 const float* __restrict__ z_a,
                 const float* __restrict__ Bmus,
                 const float* __restrict__ Blvs,
                 float* __restrict__ ws) {
    const int wave = (int)((blockIdx.x * blockDim.x + threadIdx.x) >> 5);
    const int lane = threadIdx.x & 31;
    const int n    = wave;               // 0..1023, grid sized exactly
    const int half = lane >> 4;          // 0: lanes 0-15, 1: lanes 16-31
    const int l16  = lane & 15;

    const float* __restrict__ Am = Bmus + (size_t)n * ZI_ * ZA_;
    const float* __restrict__ Al = Blvs + (size_t)n * ZI_ * ZA_;

    // accumulators: prior means / prior logvars, 4 f-tiles x 2 b-tiles
    v8f cmu[4][2] = {};
    v8f clv[4][2] = {};

    for (int k0 = 0; k0 < ZA_; k0 += 4) {
        const int k = k0 + 2 * half;     // this lane supplies K=k, k+1
        // B fragment (4x16, K x b): VGPR0 = K={k0 | k0+2}, VGPR1 = next K
        v2f bfrag[2];
        #pragma unroll
        for (int bt = 0; bt < 2; ++bt)
            bfrag[bt] = *(const v2f*)(z_a + (size_t)(bt * 16 + l16) * ZA_ + k);
        #pragma unroll
        for (int ft = 0; ft < 4; ++ft) {
            // A fragment (16x4, f x K): lane row M=l16, 8-byte NT load per lane
            const v2f am = __builtin_nontemporal_load(
                (const v2f*)(Am + (size_t)(ft * 16 + l16) * ZA_ + k));
            const v2f al = __builtin_nontemporal_load(
                (const v2f*)(Al + (size_t)(ft * 16 + l16) * ZA_ + k));
            #pragma unroll
            for (int bt = 0; bt < 2; ++bt) {
                cmu[ft][bt] = __builtin_amdgcn_wmma_f32_16x16x4_f32(
                    false, am, false, bfrag[bt], (short)0, cmu[ft][bt], false, false);
                clv[ft][bt] = __builtin_amdgcn_wmma_f32_16x16x4_f32(
                    false, al, false, bfrag[bt], (short)0, clv[ft][bt], false, false);
            }
        }
    }

    // KL epilogue. C/D layout: c[r] @ lane -> (f = ft*16 + r + 8*half, b = bt*16 + l16)
    #pragma unroll
    for (int bt = 0; bt < 2; ++bt) {
        const int b = bt * 16 + l16;
        float s1 = 0.f, s2 = 0.f, st = 0.f, si = 0.f;
        #pragma unroll
        for (int ft = 0; ft < 4; ++ft) {
            #pragma unroll
            for (int r = 0; r < 8; ++r) {
                const float pm = cmu[ft][bt][r];
                const float pl = clv[ft][bt][r];
                const int f = ft * 16 + 8 * half + r;
                const size_t idx = ((size_t)b * N_ + n) * ZI_ + f;
                const float m1 = __builtin_nontemporal_load(mu_i + idx);
                const float l1 = __builtin_nontemporal_load(logvar_i + idx);
                s1 += l1;
                s2 += pl;
                st += __expf(l1 - pl);
                const float d = pm - m1;
                si += d * d * __expf(-pl);
            }
        }
        // join lane halves: each lane then holds full sums over f=0..63 for its b
        s1 += __shfl_xor(s1, 16, 32);
        s2 += __shfl_xor(s2, 16, 32);
        st += __shfl_xor(st, 16, 32);
        si += __shfl_xor(si, 16, 32);
        float kl = 0.5f * __expf(s1 - s2) - (float)ZI_ + st + si;
        // sum the 16 distinct b's of this tile (lanes 16-31 hold duplicates)
        #pragma unroll
        for (int off = 1; off < 16; off <<= 1) kl += __shfl_xor(kl, off, 32);
        if (lane == 0) atomicAdd(ws + 2, kl);
    }
}

// ---------------- 4) finalize: 4 scalars
__global__ void finalize_kernel(const float* __restrict__ ws, float* __restrict__ out) {
    if (threadIdx.x == 0 && blockIdx.x == 0) {
        const float ar = ws[0] / (float)B_;   // sum((-1,-2)).mean()
        const float ka = ws[1] / (float)B_;
        const float ki = ws[2] / (float)B_;
        out[0] = ar;
        out[1] = ka;
        out[2] = ki;
        out[3] = -(ar + ka + ki);             // BETA_A = BETA_I = 1
    }
}

extern "C" void kernel_launch(void* const* d_in, const int* in_sizes, int n_in,
                              void* d_out, int out_size, void* d_ws, size_t ws_size,
                              hipStream_t stream) {
    const float* A_hat    = (const float*)d_in[0];
    const float* A_true   = (const float*)d_in[1];
    // d_in[2]=X_hat, d_in[3]=X_true unused (X_recon = 0)
    const float* mu_A     = (const float*)d_in[4];
    const float* logvar_A = (const float*)d_in[5];
    const float* mu_i     = (const float*)d_in[6];
    const float* logvar_i = (const float*)d_in[7];
    const float* z_a      = (const float*)d_in[8];
    const float* Bmus     = (const float*)d_in[9];
    const float* Blvs     = (const float*)d_in[10];
    float* out = (float*)d_out;
    float* ws  = (float*)d_ws;

    init_ws_kernel<<<1, 32, 0, stream>>>(ws);
    adj_recon_kernel<<<4096, 256, 0, stream>>>(A_hat, A_true, ws);
    kl_a_kernel<<<32, 128, 0, stream>>>(mu_A, logvar_A, ws);
    kl_i_kernel<<<128, 256, 0, stream>>>(mu_i, logvar_i, z_a, Bmus, Blvs, ws);
    finalize_kernel<<<1, 1, 0, stream>>>(ws, out);
}